// BatchReLUTransformer_45646912422319
// MI455X (gfx1250) — compile-verified
//
#include <hip/hip_runtime.h>

// ---------------------------------------------------------------------------
// BatchReLUTransformer forward relaxation — pure streaming elementwise kernel.
//
// Per (n,b): l,u = bounds; ll,lu = last_bounds; beta.
//   ind1 = u<=0, ind2 = l>0, ind3 = (u>0 && l<0)
//   diff   = ind3 ? u-l : 1
//   lmbda  = ind2 ? 1 : (ind3 ? u/diff : 0)
//   mu     = ind3 ? -l*(u/diff) : 0
//   beta_e = ind1 ? 0 : (ind2 ? 1 : beta)
//   cur_l  = ind2 ? l : 0
//   cur_u  = ind2 ? u : (ind3 ? u : 0)
//   new_l  = max(beta_e,0)*ll + min(beta_e,0)*lu
//   new_u  = max(lmbda,0)*lu + min(lmbda,0)*ll + mu
//   out    = { max(cur_l,new_l), min(cur_u,new_u) }
//
// Memory-bound: 28 B/element, ~470 MB/call -> ~20 us at 23.3 TB/s.
// Strategy: 128-bit coalesced loads/stores with non-temporal hints
// (stream-once working set > 192 MB L2), wave32-friendly flat indexing.
// ---------------------------------------------------------------------------

typedef float v4f __attribute__((ext_vector_type(4)));
typedef float v2f __attribute__((ext_vector_type(2)));

__device__ __forceinline__ void relax_pair(float l, float u, float beta,
                                           float ll, float lu,
                                           float& out_l, float& out_u) {
    const bool ind1 = (u <= 0.0f);
    const bool ind2 = (l > 0.0f);
    const bool ind3 = (u > 0.0f) && (l < 0.0f);

    const float diff  = ind3 ? (u - l) : 1.0f;   // safe divisor outside ind3
    const float r     = u / diff;                // == u/(u-l) when ind3
    const float lmbda = ind2 ? 1.0f : (ind3 ? r : 0.0f);
    const float mu    = ind3 ? (-l * r) : 0.0f;
    const float beff  = ind1 ? 0.0f : (ind2 ? 1.0f : beta);

    const float cur_l = ind2 ? l : 0.0f;
    const float cur_u = ind2 ? u : (ind3 ? u : 0.0f);

    const float new_l = fmaxf(beff, 0.0f) * ll + fminf(beff, 0.0f) * lu;
    const float new_u = fmaxf(lmbda, 0.0f) * lu + fminf(lmbda, 0.0f) * ll + mu;

    out_l = fmaxf(cur_l, new_l);
    out_u = fminf(cur_u, new_u);
}

// Each thread handles 2 (l,u) pairs: one b128 from bounds, one b128 from
// last_bounds, one b64 of beta, one b128 store. All accesses unit-stride
// across the wave -> fully coalesced; NT temporal hint keeps L2 clean.
__global__ __launch_bounds__(256)
void relu_relax_v4(const v4f* __restrict__ bounds,
                   const v2f* __restrict__ beta,
                   const v4f* __restrict__ last,
                   v4f* __restrict__ out,
                   long long nquads) {
    long long i = (long long)blockIdx.x * blockDim.x + threadIdx.x;
    const long long stride = (long long)gridDim.x * blockDim.x;
    for (; i < nquads; i += stride) {
        const v4f b  = __builtin_nontemporal_load(bounds + i);
        const v4f lb = __builtin_nontemporal_load(last + i);
        const v2f be = __builtin_nontemporal_load(beta + i);
        float o0, o1, o2, o3;
        relax_pair(b.x, b.y, be.x, lb.x, lb.y, o0, o1);
        relax_pair(b.z, b.w, be.y, lb.z, lb.w, o2, o3);
        v4f o;
        o.x = o0; o.y = o1; o.z = o2; o.w = o3;
        __builtin_nontemporal_store(o, out + i);
    }
}

// Scalar tail for an odd number of pairs (not hit for N*B = 16,777,216,
// but keeps kernel_launch correct for any size).
__global__ __launch_bounds__(32)
void relu_relax_tail(const float* __restrict__ bounds,
                     const float* __restrict__ beta,
                     const float* __restrict__ last,
                     float* __restrict__ out,
                     long long first_pair, long long npairs) {
    long long p = first_pair + threadIdx.x;
    if (p < npairs) {
        float ol, ou;
        relax_pair(bounds[2 * p], bounds[2 * p + 1], beta[p],
                   last[2 * p], last[2 * p + 1], ol, ou);
        out[2 * p] = ol;
        out[2 * p + 1] = ou;
    }
}

extern "C" void kernel_launch(void* const* d_in, const int* in_sizes, int n_in,
                              void* d_out, int out_size, void* d_ws, size_t ws_size,
                              hipStream_t stream) {
    (void)n_in; (void)out_size; (void)d_ws; (void)ws_size;

    const float* bounds = (const float*)d_in[0];   // (N,B,2) f32
    const float* beta   = (const float*)d_in[1];   // (N,B)   f32
    const float* last   = (const float*)d_in[2];   // (N,B,2) f32
    float* out          = (float*)d_out;           // (N,B,2) f32

    const long long npairs = (long long)in_sizes[1];  // N*B
    const long long nquads = npairs >> 1;             // 2 pairs per thread

    if (nquads > 0) {
        const int block = 256;
        long long blocks = (nquads + block - 1) / block;
        if (blocks > 2147483647LL) blocks = 2147483647LL;  // grid-stride covers rest
        relu_relax_v4<<<(unsigned)blocks, block, 0, stream>>>(
            (const v4f*)bounds, (const v2f*)beta, (const v4f*)last,
            (v4f*)out, nquads);
    }
    if (npairs & 1LL) {
        relu_relax_tail<<<1, 32, 0, stream>>>(bounds, beta, last, out,
                                              nquads * 2, npairs);
    }
}